// CosineSimCodebook_51049981281495
// MI455X (gfx1250) — compile-verified
//
#include <hip/hip_runtime.h>
#include <stdint.h>

typedef float v2f __attribute__((ext_vector_type(2)));
typedef float v8f __attribute__((ext_vector_type(8)));
typedef int   v4i __attribute__((vector_size(16)));   // matches builtin param pointee

#define DECAY      0.8f
#define OMDECAY    0.2f
#define EPSF       1e-5f

// Problem dims
#define M_TOT 16384      // B*N tokens
#define K_TOT 512        // D
#define C_TOT 4096       // codes

// GEMM tiling
#define BM     128
#define BN     128
#define KT     16
#define LDSS   20        // padded LDS row stride (floats): conflict-free, 16B aligned
#define NCHUNK (K_TOT / KT)

// Output offsets (floats, concatenated in reference return order)
#define OUT_QUANT  0
#define OUT_IND    8388608
#define OUT_DIST   8404992
#define OUT_NEMBED 75513856
#define OUT_NCS    77611008
#define OUT_NEA    77615104

// Workspace layout (bytes)
#define WS_ARGMAX  0
#define WS_BINS    (M_TOT * 8)                   // after 16384 u64
#define WS_ESUM    (WS_BINS + C_TOT * 4)
#define WS_TOTAL   (WS_ESUM + C_TOT * K_TOT * 4)
#define WS_BYTES   (WS_TOTAL + 4)

#if defined(__has_builtin)
#if __has_builtin(__builtin_amdgcn_global_load_async_to_lds_b128) && \
    __has_builtin(__builtin_amdgcn_s_wait_asynccnt)
#define USE_ASYNC_LDS 1
#endif
#endif
#ifndef USE_ASYNC_LDS
#define USE_ASYNC_LDS 0
#endif

#if USE_ASYNC_LDS
#define GPTR(p) ((__attribute__((address_space(1))) v4i*)(p))
#define SPTR(p) ((__attribute__((address_space(3))) v4i*)(p))
#endif

__global__ __launch_bounds__(256)
void k_zero_ws(uint32_t* __restrict__ ws, int nwords) {
    int i = blockIdx.x * blockDim.x + threadIdx.x;
    if (i < nwords) ws[i] = 0u;
}

// dist = x(16384x512) * embed^T(512x4096), fused per-row argmax via packed u64 atomic max.
// Double-buffered LDS pipeline; async global->LDS copies when available.
__global__ __launch_bounds__(256)
void k_gemm_dist(const float* __restrict__ x, const float* __restrict__ embed,
                 float* __restrict__ dist, unsigned long long* __restrict__ argmax) {
    __shared__ float As[2 * BM * LDSS];
    __shared__ float Bs[2 * BN * LDSS];

    const int tid    = threadIdx.x;
    const int lane   = tid & 31;
    const int wave   = tid >> 5;
    const int wm     = wave >> 1;      // 0..3 : 32-row group
    const int wn     = wave & 1;       // 0..1 : 64-col group
    const int lane15 = lane & 15;
    const int khalf  = lane >> 4;      // fragments hold K pair 2*khalf, 2*khalf+1
    const int rowBase = blockIdx.y * BM;
    const int colBase = blockIdx.x * BN;

    // Staging coords: each thread moves 2 float4 per tile (A and B) per chunk.
    const int r0 = tid >> 2;           // 0..63   (second slice: r0+64)
    const int c0 = (tid & 3) << 2;     // 0,4,8,12

    const float* gx0 = &x[(size_t)(rowBase + r0) * K_TOT + c0];
    const float* gx1 = gx0 + (size_t)64 * K_TOT;
    const float* ge0 = &embed[(size_t)(colBase + r0) * K_TOT + c0];
    const float* ge1 = ge0 + (size_t)64 * K_TOT;
    const int lo0 = r0 * LDSS + c0;
    const int lo1 = lo0 + 64 * LDSS;

#if USE_ASYNC_LDS
    auto issue = [&](int kb, int buf) {
        float* la = &As[buf * BM * LDSS];
        float* lb = &Bs[buf * BN * LDSS];
        __builtin_amdgcn_global_load_async_to_lds_b128(
            GPTR(gx0 + kb), SPTR(&la[lo0]), 0, 0);
        __builtin_amdgcn_global_load_async_to_lds_b128(
            GPTR(gx1 + kb), SPTR(&la[lo1]), 0, 0);
        __builtin_amdgcn_global_load_async_to_lds_b128(
            GPTR(ge0 + kb), SPTR(&lb[lo0]), 0, 0);
        __builtin_amdgcn_global_load_async_to_lds_b128(
            GPTR(ge1 + kb), SPTR(&lb[lo1]), 0, 0);
    };
#else
    float4 pa0, pa1, pb0, pb1;
    auto load_regs = [&](int kb) {
        pa0 = *(const float4*)(gx0 + kb);
        pa1 = *(const float4*)(gx1 + kb);
        pb0 = *(const float4*)(ge0 + kb);
        pb1 = *(const float4*)(ge1 + kb);
    };
    auto store_regs = [&](int buf) {
        float* la = &As[buf * BM * LDSS];
        float* lb = &Bs[buf * BN * LDSS];
        *(float4*)&la[lo0] = pa0;
        *(float4*)&la[lo1] = pa1;
        *(float4*)&lb[lo0] = pb0;
        *(float4*)&lb[lo1] = pb1;
    };
#endif

    v8f acc[2][4];
#pragma unroll
    for (int i = 0; i < 2; ++i)
#pragma unroll
        for (int j = 0; j < 4; ++j)
            acc[i][j] = (v8f){};

    // Prologue: start chunk 0
#if USE_ASYNC_LDS
    issue(0, 0);
#else
    load_regs(0);
#endif

    for (int chunk = 0; chunk < NCHUNK; ++chunk) {
        const int kb  = chunk * KT;
        const int buf = chunk & 1;
        const bool more = (chunk + 1) < NCHUNK;
        __syncthreads();   // all waves done reading buf^1 (previous chunk's compute)
#if USE_ASYNC_LDS
        if (more) {
            issue(kb + KT, buf ^ 1);
            __builtin_amdgcn_s_wait_asynccnt(4);   // this chunk's 4 ops landed
        } else {
            __builtin_amdgcn_s_wait_asynccnt(0);
        }
#else
        store_regs(buf);                 // stage this chunk into LDS
        if (more) load_regs(kb + KT);    // next chunk's global loads fly during compute
#endif
        __syncthreads();   // everyone's data for this chunk is in LDS

        const float* __restrict__ Ab = &As[buf * BM * LDSS];
        const float* __restrict__ Bb = &Bs[buf * BN * LDSS];
#pragma unroll
        for (int k0 = 0; k0 < KT; k0 += 4) {
            const int kk = k0 + 2 * khalf;
            v2f af[2], bf[4];
#pragma unroll
            for (int i = 0; i < 2; ++i)
                af[i] = *(const v2f*)&Ab[(wm * 32 + i * 16 + lane15) * LDSS + kk];
#pragma unroll
            for (int j = 0; j < 4; ++j)
                bf[j] = *(const v2f*)&Bb[(wn * 64 + j * 16 + lane15) * LDSS + kk];
#pragma unroll
            for (int i = 0; i < 2; ++i)
#pragma unroll
                for (int j = 0; j < 4; ++j)
                    acc[i][j] = __builtin_amdgcn_wmma_f32_16x16x4_f32(
                        false, af[i], false, bf[j], (short)0, acc[i][j], false, false);
        }
    }

    // Epilogue: NT-store dist, fused packed argmax (val desc, col asc on ties)
#pragma unroll
    for (int i = 0; i < 2; ++i) {
#pragma unroll
        for (int v = 0; v < 8; ++v) {
            const int rowG = rowBase + wm * 32 + i * 16 + v + 8 * khalf;
            unsigned long long best = 0ull;
#pragma unroll
            for (int j = 0; j < 4; ++j) {
                const int colG = colBase + wn * 64 + j * 16 + lane15;
                const float val = acc[i][j][v];
                __builtin_nontemporal_store(val, &dist[(size_t)rowG * C_TOT + colG]);
                const unsigned b   = __float_as_uint(val);
                const unsigned key = (b & 0x80000000u) ? ~b : (b | 0x80000000u);
                const unsigned long long p =
                    ((unsigned long long)key << 32) |
                    (unsigned long long)(0xFFFFFFFFu - (unsigned)colG);
                best = (p > best) ? p : best;
            }
            // reduce across the 16-lane half that shares this row
#pragma unroll
            for (int mask = 1; mask < 16; mask <<= 1) {
                unsigned lo = (unsigned)best;
                unsigned hi = (unsigned)(best >> 32);
                lo = __shfl_xor(lo, mask, 32);
                hi = __shfl_xor(hi, mask, 32);
                const unsigned long long o = ((unsigned long long)hi << 32) | lo;
                best = (o > best) ? o : best;
            }
            if (lane15 == 0)
                atomicMax(&argmax[rowG], best);
        }
    }
}

// Per token: decode argmax, write index, gather quantize row, bin count, scatter embed_sum.
__global__ __launch_bounds__(128)
void k_assign(const float* __restrict__ x, const float* __restrict__ embed,
              const unsigned long long* __restrict__ argmax,
              float* __restrict__ quant, float* __restrict__ ind_out,
              float* __restrict__ bins, float* __restrict__ esum) {
    const int row = blockIdx.x;
    const int tid = threadIdx.x;
    const unsigned long long p = argmax[row];
    const unsigned ind = 0xFFFFFFFFu - (unsigned)(p & 0xFFFFFFFFull);
    if (tid == 0) {
        ind_out[row] = (float)ind;
        atomicAdd(&bins[ind], 1.0f);
    }
    const float4 e = *(const float4*)&embed[(size_t)ind * K_TOT + tid * 4];
    *(float4*)&quant[(size_t)row * K_TOT + tid * 4] = e;
    const float4 xv = *(const float4*)&x[(size_t)row * K_TOT + tid * 4];
    float* es = &esum[(size_t)ind * K_TOT + tid * 4];
    atomicAdd(es + 0, xv.x);
    atomicAdd(es + 1, xv.y);
    atomicAdd(es + 2, xv.z);
    atomicAdd(es + 3, xv.w);
}

// EMA cluster-size update + total reduction (single block, 4096 elems).
__global__ __launch_bounds__(1024)
void k_cluster(const float* __restrict__ cs, const float* __restrict__ bins,
               float* __restrict__ ncs_out, float* __restrict__ total_out) {
    __shared__ float red[1024];
    const int tid = threadIdx.x;
    float s = 0.f;
#pragma unroll
    for (int k = 0; k < 4; ++k) {
        const int c = tid + k * 1024;
        const float v = cs[c] * DECAY + bins[c] * OMDECAY;
        ncs_out[c] = v;
        s += v;
    }
    red[tid] = s;
    __syncthreads();
    for (int off = 512; off > 0; off >>= 1) {
        if (tid < off) red[tid] += red[tid + off];
        __syncthreads();
    }
    if (tid == 0) total_out[0] = red[0];
}

// Per code: new_embed_avg EMA, laplace-smoothed scale, l2norm -> new_embed.
__global__ __launch_bounds__(128)
void k_embed_update(const float* __restrict__ ea, const float* __restrict__ esum,
                    const float* __restrict__ ncs, const float* __restrict__ total_ws,
                    float* __restrict__ nea_out, float* __restrict__ nembed_out) {
    __shared__ float red[128];
    const int c   = blockIdx.x;
    const int tid = threadIdx.x;
    const float total    = total_ws[0];
    const float smoothed = (ncs[c] + EPSF) / (total + (float)C_TOT * EPSF) * total;
    const float inv_s    = 1.0f / smoothed;

    const float4 a  = *(const float4*)&ea[(size_t)c * K_TOT + tid * 4];
    const float4 su = *(const float4*)&esum[(size_t)c * K_TOT + tid * 4];
    float4 v;
    v.x = a.x * DECAY + su.x * OMDECAY;
    v.y = a.y * DECAY + su.y * OMDECAY;
    v.z = a.z * DECAY + su.z * OMDECAY;
    v.w = a.w * DECAY + su.w * OMDECAY;
    *(float4*)&nea_out[(size_t)c * K_TOT + tid * 4] = v;

    float4 w;
    w.x = v.x * inv_s; w.y = v.y * inv_s; w.z = v.z * inv_s; w.w = v.w * inv_s;
    red[tid] = w.x * w.x + w.y * w.y + w.z * w.z + w.w * w.w;
    __syncthreads();
    for (int off = 64; off > 0; off >>= 1) {
        if (tid < off) red[tid] += red[tid + off];
        __syncthreads();
    }
    const float norm = sqrtf(red[0]);
    const float invd = 1.0f / fmaxf(norm, 1e-12f);
    float4 o;
    o.x = w.x * invd; o.y = w.y * invd; o.z = w.z * invd; o.w = w.w * invd;
    *(float4*)&nembed_out[(size_t)c * K_TOT + tid * 4] = o;
}

extern "C" void kernel_launch(void* const* d_in, const int* in_sizes, int n_in,
                              void* d_out, int out_size, void* d_ws, size_t ws_size,
                              hipStream_t stream) {
    const float* x     = (const float*)d_in[0];   // (16,1024,512)
    const float* embed = (const float*)d_in[1];   // (1,4096,512)
    const float* cs    = (const float*)d_in[2];   // (1,4096)
    const float* ea    = (const float*)d_in[3];   // (1,4096,512)

    float* out     = (float*)d_out;
    float* quant   = out + OUT_QUANT;
    float* ind_out = out + OUT_IND;
    float* dist    = out + OUT_DIST;
    float* nembed  = out + OUT_NEMBED;
    float* ncs     = out + OUT_NCS;
    float* nea     = out + OUT_NEA;

    char* ws = (char*)d_ws;
    unsigned long long* argmax = (unsigned long long*)(ws + WS_ARGMAX);
    float* bins  = (float*)(ws + WS_BINS);
    float* esum  = (float*)(ws + WS_ESUM);
    float* total = (float*)(ws + WS_TOTAL);

    const int nwords = WS_BYTES / 4;
    k_zero_ws<<<(nwords + 255) / 256, 256, 0, stream>>>((uint32_t*)ws, nwords);

    dim3 grid(C_TOT / BN, M_TOT / BM);   // (32, 128)
    k_gemm_dist<<<grid, 256, 0, stream>>>(x, embed, dist, argmax);

    k_assign<<<M_TOT, 128, 0, stream>>>(x, embed, argmax, quant, ind_out, bins, esum);

    k_cluster<<<1, 1024, 0, stream>>>(cs, bins, ncs, total);

    k_embed_update<<<C_TOT, 128, 0, stream>>>(ea, esum, ncs, total, nea, nembed);
}